// CVQVAECodebook_65051574665891
// MI455X (gfx1250) — compile-verified
//
#include <hip/hip_runtime.h>

// CDNA5 / gfx1250, wave32. VQ-VAE codebook assignment.
// argmin_k ||x||^2 + ||e_k||^2 - 2 x.e_k  ==  argmin_k ||e_k||^2 - 2 x.e_k
// (||x||^2 is constant per row -> dropped).

typedef float v2f __attribute__((ext_vector_type(2)));
typedef float v4f __attribute__((ext_vector_type(4)));
typedef float v8f __attribute__((ext_vector_type(8)));

#define N_ROWS  131072
#define K_CODES 1024
#define D_DIM   64

// ---------------------------------------------------------------------------
// Kernel 1: code norms ||e_k||^2  (1024 values -> d_ws)
// ---------------------------------------------------------------------------
__global__ __launch_bounds__(256) void vq_norms_kernel(
    const float* __restrict__ emb, float* __restrict__ norms)
{
    int k = blockIdx.x * 256 + threadIdx.x;           // 4 blocks x 256 = 1024
    const v4f* row = (const v4f*)(emb + (size_t)k * D_DIM);
    float s = 0.0f;
    #pragma unroll
    for (int j = 0; j < D_DIM / 4; ++j) {
        v4f v = row[j];
        s += v.x * v.x + v.y * v.y + v.z * v.z + v.w * v.w;
    }
    norms[k] = s;
}

// ---------------------------------------------------------------------------
// Kernel 2: main. Block = 256 threads = 8 waves; each wave owns 16 rows of x.
// Codebook streamed through double-buffered LDS via GLOBAL_LOAD_ASYNC_TO_LDS
// (ASYNCcnt), overlapped with the v_wmma_f32_16x16x4_f32 inner loop.
// ---------------------------------------------------------------------------
__global__ __launch_bounds__(256) void vq_assign_kernel(
    const float* __restrict__ x,
    const float* __restrict__ emb,
    const float* __restrict__ norms,
    float* __restrict__ zq,          // [N, 64]
    float* __restrict__ probs)       // [N, 1024]
{
    __shared__ float lds_norm[K_CODES];          // 4 KB
    __shared__ float lds_e[2][16 * D_DIM];       // 2 x 4 KB double buffer

    const int tid  = threadIdx.x;
    const int lane = tid & 31;
    const int wave = tid >> 5;
    const int half = lane >> 4;                  // 0 or 1 (wave32 halves)
    const int l16  = lane & 15;

    // stage the 1024 code norms once (normal path; covered by first barrier)
    for (int i = tid; i < K_CODES; i += 256) lds_norm[i] = norms[i];

    const int r0 = blockIdx.x * 128 + wave * 16; // first row of this wave's tile

    // --- load A fragments: 16 K-chunks of 4, f32 WMMA A layout ---
    // lane half h holds K = 4*kc + 2h .. +1 for row M = lane%16
    v2f a[16];
    {
        const float* xrow = x + (size_t)(r0 + l16) * D_DIM;
        #pragma unroll
        for (int kc = 0; kc < 16; ++kc) {
            a[kc] = *(const v2f*)(xrow + kc * 4 + half * 2);
        }
    }

    float minv[8];
    int   mini[8];
    #pragma unroll
    for (int v = 0; v < 8; ++v) { minv[v] = 3.4e38f; mini[v] = 0; }

    // Async-stage one 16-code tile (4KB): each thread moves 16B mem->LDS.
    // One instruction per wave per tile -> ASYNCcnt tracks <=2 outstanding.
    auto stage_async = [&](int n0, int buf) {
        unsigned dst = (unsigned)(size_t)(&lds_e[buf][0]) + (unsigned)(tid * 16);
        const float* src = emb + (size_t)n0 * D_DIM + tid * 4;
        asm volatile("global_load_async_to_lds_b128 %0, %1, off"
                     :: "v"(dst), "v"(src) : "memory");
    };

    stage_async(0, 0);                           // prologue: tile 0 -> buf 0

    // --- sweep the 1024 codes in 16-wide column tiles, double-buffered ---
    for (int t = 0; t < K_CODES / 16; ++t) {
        if (t + 1 < K_CODES / 16) {
            // buf[(t+1)&1] was released by the barrier ending iteration t-1
            stage_async((t + 1) * 16, (t + 1) & 1);
            asm volatile("s_wait_asynccnt 0x1" ::: "memory"); // tile t landed
        } else {
            asm volatile("s_wait_asynccnt 0x0" ::: "memory");
        }
        __syncthreads();                         // all waves' portions visible

        const float* e = lds_e[t & 1];
        v8f acc = {0.f, 0.f, 0.f, 0.f, 0.f, 0.f, 0.f, 0.f};
        #pragma unroll
        for (int kc = 0; kc < 16; ++kc) {
            // B (4x16): lane holds code n0 + lane%16, K = 4*kc + 2*half .. +1
            v2f b = *(const v2f*)(&e[l16 * D_DIM + kc * 4 + half * 2]);
            acc = __builtin_amdgcn_wmma_f32_16x16x4_f32(
                false, a[kc], false, b, (short)0, acc, false, false);
        }

        // distance (up to per-row constant): ||e||^2 - 2 * dot
        // nidx strictly increases per lane -> strict '<' keeps first occurrence
        const int   n0   = t * 16;
        const float nrm  = lds_norm[n0 + l16];
        const int   nidx = n0 + l16;
        #pragma unroll
        for (int v = 0; v < 8; ++v) {
            float d = __builtin_fmaf(-2.0f, acc[v], nrm);
            if (d < minv[v]) { minv[v] = d; mini[v] = nidx; }
        }
        __syncthreads();                         // release buf[t&1] for t+2
    }

    // --- per-row argmin: reduce across the 16 lanes of each half ---
    // C/D layout: VGPR v, half h -> row M = v + 8h ; lane%16 = code subset.
    // Cross-lane mixing needs the index tie-break (first-occurrence argmin).
    #pragma unroll
    for (int v = 0; v < 8; ++v) {
        float bv = minv[v]; int bi = mini[v];
        #pragma unroll
        for (int off = 1; off < 16; off <<= 1) {
            float ov = __shfl_xor(bv, off, 32);
            int   oi = __shfl_xor(bi, off, 32);
            if (ov < bv || (ov == bv && oi < bi)) { bv = ov; bi = oi; }
        }
        minv[v] = bv; mini[v] = bi;
    }

    // --- write z_q (gather) and one-hot probs; streaming non-temporal stores
    #pragma unroll 1
    for (int r = 0; r < 16; ++r) {
        const int v = r & 7, h = r >> 3;
        const int idx = __shfl(mini[v], h * 16, 32);   // broadcast to wave

        // z_q row: 64 floats, 2 per lane (codebook row is L2-hot)
        v2f e2 = ((const v2f*)(emb + (size_t)idx * D_DIM))[lane];
        __builtin_nontemporal_store(
            e2, (v2f*)(zq + (size_t)(r0 + r) * D_DIM) + lane);

        // probs row: 1024 floats as 8 x (32 lanes x b128), NT (write-once)
        v4f* prow = (v4f*)(probs + (size_t)(r0 + r) * K_CODES);
        #pragma unroll
        for (int it = 0; it < 8; ++it) {
            const int c = (it * 32 + lane) * 4;
            v4f pv;
            pv.x = (c + 0 == idx) ? 1.0f : 0.0f;
            pv.y = (c + 1 == idx) ? 1.0f : 0.0f;
            pv.z = (c + 2 == idx) ? 1.0f : 0.0f;
            pv.w = (c + 3 == idx) ? 1.0f : 0.0f;
            __builtin_nontemporal_store(pv, prow + it * 32 + lane);
        }
    }
}

// ---------------------------------------------------------------------------
extern "C" void kernel_launch(void* const* d_in, const int* in_sizes, int n_in,
                              void* d_out, int out_size, void* d_ws, size_t ws_size,
                              hipStream_t stream) {
    const float* x   = (const float*)d_in[0];   // [N, 64]
    const float* emb = (const float*)d_in[1];   // [1024, 64]
    float* norms = (float*)d_ws;                // 1024 f32 scratch
    float* zq    = (float*)d_out;                           // [N, 64]
    float* probs = (float*)d_out + (size_t)N_ROWS * D_DIM;  // [N, 1024]

    vq_norms_kernel<<<K_CODES / 256, 256, 0, stream>>>(emb, norms);
    vq_assign_kernel<<<N_ROWS / 128, 256, 0, stream>>>(x, emb, norms, zq, probs);
}